// ScaledDotProductAttention_77627238908122
// MI455X (gfx1250) — compile-verified
//
#include <hip/hip_runtime.h>
#include <hip/hip_bf16.h>
#include <math.h>

typedef __attribute__((ext_vector_type(2))) float v2f;
typedef __attribute__((ext_vector_type(8))) float v8f;

#define BATCH 16
#define SEQ   2048
#define DIM   64
#define QTILE 16
#define NWAVES 8
#define INV_TEMP 0.125f   // 1/sqrt(64)

// One workgroup: batch b = blockIdx.y, q-rows [q0, q0+16) with q0 = blockIdx.x*16.
// 256 threads = 8 wave32.
__global__ __launch_bounds__(256)
void ScaledDotProductAttention_77627238908122_kernel(
    const float* __restrict__ q,
    const float* __restrict__ k,
    const float* __restrict__ v,
    const unsigned char* __restrict__ mask,
    float* __restrict__ out,      // [B,S,D]
    float* __restrict__ attn)     // [B,S,S]
{
    __shared__ float s_scores[QTILE * SEQ];   // 128 KB: raw scores, then probs
    __shared__ float s_out[QTILE * DIM];      // 4 KB: cross-wave PV reduction
    __shared__ float s_rmax[QTILE];
    __shared__ float s_rinv[QTILE];

    const int b  = blockIdx.y;
    const int q0 = blockIdx.x * QTILE;

    const int lane = threadIdx.x & 31;
    const int wv   = threadIdx.x >> 5;
    const int mrow = lane & 15;           // M (or N) index within a 16-wide tile
    const int hsel = (lane >> 4) & 1;     // 0: lanes 0-15 (K=0,1) ; 1: lanes 16-31 (K=2,3)

    const float* qb = q + (size_t)b * SEQ * DIM;
    const float* kb = k + (size_t)b * SEQ * DIM;
    const float* vb = v + (size_t)b * SEQ * DIM;
    const unsigned char* mb = mask + (size_t)b * SEQ * SEQ;

    // Zero the PV reduction tile early (barriers before its use come later).
    for (int i = threadIdx.x; i < QTILE * DIM; i += 256) s_out[i] = 0.0f;

    // ---- Preload Q A-fragments (16x4 fp32 per WMMA step, 16 steps over D=64) ----
    // A layout (ISA 7.12.2): lane(0-15)=M, VGPR0 holds K=0 (lanes<16) / K=2 (lanes>=16),
    // VGPR1 holds K=1 / K=3  ->  each lane loads the consecutive pair d = 4*step + 2*hsel {+0,+1}.
    v2f qa[16];
    {
        const float* qp = qb + (size_t)(q0 + mrow) * DIM + 2 * hsel;
#pragma unroll
        for (int ks = 0; ks < 16; ++ks)
            qa[ks] = *(const v2f*)(qp + 4 * ks);
    }

    // ================= Pass 1: scores = mask ? -inf : (Q K^T)/8 -> LDS =================
    // Wave wv handles 16 column tiles; two per iteration (j, j+64) so the two
    // WMMA accumulator chains interleave instead of one serial RAW chain.
    for (int j = wv; j < 64; j += NWAVES) {
        const int jB = j + 64;
        v8f accA = {};
        v8f accB = {};
        const float* kpA = kb + (size_t)(j  * 16 + mrow) * DIM + 2 * hsel;
        const float* kpB = kb + (size_t)(jB * 16 + mrow) * DIM + 2 * hsel;
#pragma unroll
        for (int ks = 0; ks < 16; ++ks) {
            // B (4x16) layout mirrors A: VGPR0 = rows K={0,2}, VGPR1 = rows K={1,3};
            // B[kd][n] = K[j*16+n][d] -> consecutive d-pair per lane.
            v2f bA = *(const v2f*)(kpA + 4 * ks);
            v2f bB = *(const v2f*)(kpB + 4 * ks);
            accA = __builtin_amdgcn_wmma_f32_16x16x4_f32(
                false, qa[ks], false, bA, (short)0, accA, false, false);
            accB = __builtin_amdgcn_wmma_f32_16x16x4_f32(
                false, qa[ks], false, bB, (short)0, accB, false, false);
        }
        // D layout: lane l, VGPR r -> element (M = r + 8*hsel, N = lane&15)
        const int colA = j  * 16 + mrow;
        const int colB = jB * 16 + mrow;
#pragma unroll
        for (int r = 0; r < 8; ++r) {
            const int m = r + 8 * hsel;
            const unsigned char mA = mb[(size_t)(q0 + m) * SEQ + colA];
            const unsigned char mB = mb[(size_t)(q0 + m) * SEQ + colB];
            s_scores[m * SEQ + colA] = mA ? -INFINITY : accA[r] * INV_TEMP;
            s_scores[m * SEQ + colB] = mB ? -INFINITY : accB[r] * INV_TEMP;
        }
    }
    __syncthreads();

    // ================= Pass 2: per-row max and 1/sum(exp) =================
    // Wave wv reduces rows 2*wv and 2*wv+1 with wave32 shuffles.
#pragma unroll
    for (int rr = 0; rr < 2; ++rr) {
        const int r = wv * 2 + rr;
        float mx = -INFINITY;
        for (int i = lane; i < SEQ; i += 32) mx = fmaxf(mx, s_scores[r * SEQ + i]);
#pragma unroll
        for (int off = 16; off > 0; off >>= 1) mx = fmaxf(mx, __shfl_xor(mx, off, 32));
        float sum = 0.0f;
        for (int i = lane; i < SEQ; i += 32) sum += __expf(s_scores[r * SEQ + i] - mx);
#pragma unroll
        for (int off = 16; off > 0; off >>= 1) sum += __shfl_xor(sum, off, 32);
        if (lane == 0) { s_rmax[r] = mx; s_rinv[r] = 1.0f / sum; }
    }
    __syncthreads();

    // ================= Pass 3: probabilities -> LDS (in place) + global attn =================
    {
        float* ab = attn + (size_t)b * SEQ * SEQ + (size_t)q0 * SEQ;
        for (int i = threadIdx.x; i < QTILE * SEQ; i += 256) {
            const int row = i >> 11;          // /2048
            const int col = i & (SEQ - 1);
            const float p = __expf(s_scores[i] - s_rmax[row]) * s_rinv[row];
            s_scores[i] = p;
            ab[(size_t)row * SEQ + col] = p;  // coalesced 1KB bursts
        }
    }
    __syncthreads();

    // ================= Pass 4: out = P @ V =================
    // Wave wv owns K-chunk [wv*256, wv*256+256); 4 N-tiles cover D=64.
    // Even/odd K-step accumulator sets -> 8 independent WMMA chains.
    {
        v8f oaccA[4] = {};
        v8f oaccB[4] = {};
        const int kbase = wv * 256;
        for (int kk = 0; kk < 64; kk += 2) {
            const int k0 = kbase + kk * 4;
            // A-frags: P[m = lane&15][k + 2*hsel + {0,1}] from LDS (8B aligned)
            v2f aA = *(const v2f*)&s_scores[mrow * SEQ + k0 + 2 * hsel];
            v2f aB = *(const v2f*)&s_scores[mrow * SEQ + k0 + 4 + 2 * hsel];
#pragma unroll
            for (int nt = 0; nt < 4; ++nt) {
                const float* vpA = vb + (size_t)(k0     + 2 * hsel) * DIM + nt * 16 + mrow;
                const float* vpB = vb + (size_t)(k0 + 4 + 2 * hsel) * DIM + nt * 16 + mrow;
                v2f bA, bB;
                bA.x = vpA[0];      // V[k  +2*hsel  ][n]
                bA.y = vpA[DIM];    // V[k  +2*hsel+1][n]
                bB.x = vpB[0];
                bB.y = vpB[DIM];
                oaccA[nt] = __builtin_amdgcn_wmma_f32_16x16x4_f32(
                    false, aA, false, bA, (short)0, oaccA[nt], false, false);
                oaccB[nt] = __builtin_amdgcn_wmma_f32_16x16x4_f32(
                    false, aB, false, bB, (short)0, oaccB[nt], false, false);
            }
        }
        // Cross-wave reduction into s_out via LDS float atomics (ds_add_f32 path).
#pragma unroll
        for (int nt = 0; nt < 4; ++nt)
#pragma unroll
            for (int r = 0; r < 8; ++r) {
                const int m = r + 8 * hsel;
                atomicAdd(&s_out[m * DIM + nt * 16 + mrow], oaccA[nt][r] + oaccB[nt][r]);
            }
    }
    __syncthreads();

    // Write output tile (coalesced).
    {
        float* ob = out + (size_t)b * SEQ * DIM + (size_t)q0 * DIM;
        for (int i = threadIdx.x; i < QTILE * DIM; i += 256)
            ob[i] = s_out[i];
    }
}

extern "C" void kernel_launch(void* const* d_in, const int* in_sizes, int n_in,
                              void* d_out, int out_size, void* d_ws, size_t ws_size,
                              hipStream_t stream) {
    (void)in_sizes; (void)n_in; (void)d_ws; (void)ws_size; (void)out_size;
    const float* q = (const float*)d_in[0];
    const float* k = (const float*)d_in[1];
    const float* v = (const float*)d_in[2];
    const unsigned char* mask = (const unsigned char*)d_in[3];  // jax bool: 1 byte/elem

    float* out  = (float*)d_out;                                // [B,S,D] first
    float* attn = out + (size_t)BATCH * SEQ * DIM;               // then [B,S,S]

    dim3 grid(SEQ / QTILE, BATCH);   // 128 q-tiles x 16 batch-heads
    dim3 block(256);                 // 8 wave32
    ScaledDotProductAttention_77627238908122_kernel<<<grid, block, 0, stream>>>(
        q, k, v, mask, out, attn);
}